// GATAggregator_38190849196690
// MI455X (gfx1250) — compile-verified
//
#include <hip/hip_runtime.h>
#include <hip/hip_bf16.h>

typedef float v2f __attribute__((ext_vector_type(2)));
typedef float v8f __attribute__((ext_vector_type(8)));

#define NNODES 50000
#define NEDGES 500000
#define HD 128
#define NH 3
#define FDIM 384   /* NH*HD */
#define BBATCH 512
#define LSEQ 10
#define NEG_SLOPE 0.2f

__device__ __forceinline__ int f2ord(float f) {
  int i = __float_as_int(f);
  return i >= 0 ? i : (i ^ 0x7FFFFFFF);
}
__device__ __forceinline__ float ord2f(int i) {
  return __int_as_float(i >= 0 ? i : (i ^ 0x7FFFFFFF));
}

// ---------------------------------------------------------------------------
// feat[N x 384] = x[N x 128] @ W[128 x 384]
// One block per 16-row tile of x (staged in LDS via b128). 8 waves; each wave
// owns three 16-wide column tiles; K=128 runs as 32 chained v_wmma_f32_16x16x4_f32.
// A layout (16x4 fp32): lanes 0-15 hold M=lane {K=k0,k0+1}; lanes 16-31 {k0+2,k0+3}.
// B layout (4x16 fp32): VGPR0 = rows k0 (lanes 0-15) / k0+2 (lanes 16-31), VGPR1 = k0+1 / k0+3.
// C/D layout (16x16 fp32): VGPR r = row r (lanes 0-15) / row r+8 (lanes 16-31).
// ---------------------------------------------------------------------------
__global__ __launch_bounds__(256) void gemm_wmma_kernel(const float* __restrict__ x,
                                                        const float* __restrict__ W,
                                                        float* __restrict__ feat) {
  __shared__ float Atile[16][132];   // pad 128->132 (33 float4s) to spread LDS banks
  const int m0 = blockIdx.x * 16;
  const float4* x4 = (const float4*)x;
  for (int i = threadIdx.x; i < 16 * 32; i += 256) {   // 512 float4 loads
    int r = i >> 5, c = i & 31;
    *(float4*)&Atile[r][c * 4] = x4[(size_t)(m0 + r) * 32 + c];
  }
  __syncthreads();

  const int wave = threadIdx.x >> 5;
  const int lane = threadIdx.x & 31;
  const int hi   = lane >> 4;        // 0: K low pair, 1: K high pair
  const int l    = lane & 15;

  for (int t = 0; t < 3; ++t) {
    const int nb = (wave * 3 + t) * 16;    // output column base (0..368)
    v8f c = {0.f, 0.f, 0.f, 0.f, 0.f, 0.f, 0.f, 0.f};
    #pragma unroll 4
    for (int k0 = 0; k0 < HD; k0 += 4) {
      const int ka = k0 + hi * 2;
      v2f a, b;
      a.x = Atile[l][ka];
      a.y = Atile[l][ka + 1];
      const float* Wp = W + (size_t)ka * FDIM + nb + l;
      b.x = Wp[0];
      b.y = Wp[FDIM];
      c = __builtin_amdgcn_wmma_f32_16x16x4_f32(false, a, false, b, (short)0, c,
                                                false, false);
    }
    #pragma unroll
    for (int r = 0; r < 8; ++r) {
      feat[(size_t)(m0 + r + hi * 8) * FDIM + nb + l] = c[r];
    }
  }
}

// ---------------------------------------------------------------------------
// el[n][h] = dot(feat[n, h*128 : (h+1)*128], attn_l[h]);  er likewise.
// One wave per node; one float4 per lane per head; wave32 shuffle reduction.
// ---------------------------------------------------------------------------
__global__ __launch_bounds__(128) void attn_dots_kernel(const float* __restrict__ feat,
                                                        const float* __restrict__ attn_l,
                                                        const float* __restrict__ attn_r,
                                                        float* __restrict__ el,
                                                        float* __restrict__ er) {
  const int node = blockIdx.x * 4 + (threadIdx.x >> 5);
  if (node >= NNODES) return;
  const int lane = threadIdx.x & 31;
  const float4* f4  = (const float4*)(feat + (size_t)node * FDIM);
  const float4* al4 = (const float4*)attn_l;
  const float4* ar4 = (const float4*)attn_r;
  #pragma unroll
  for (int h = 0; h < NH; ++h) {
    const float4 v = f4[h * 32 + lane];
    const float4 a = al4[h * 32 + lane];
    const float4 r = ar4[h * 32 + lane];
    float sl = v.x * a.x + v.y * a.y + v.z * a.z + v.w * a.w;
    float sr = v.x * r.x + v.y * r.y + v.z * r.z + v.w * r.w;
    #pragma unroll
    for (int off = 16; off > 0; off >>= 1) {
      sl += __shfl_xor(sl, off, 32);
      sr += __shfl_xor(sr, off, 32);
    }
    if (lane == 0) {
      el[node * NH + h] = sl;
      er[node * NH + h] = sr;
    }
  }
}

// ---------------------------------------------------------------------------
// Per-layer scratch init: rst = 0, emax = -inf (ordered-int), denom = 0.
// ---------------------------------------------------------------------------
__global__ void layer_init_kernel(float* __restrict__ rst, int* __restrict__ emax,
                                  float* __restrict__ denom) {
  const int idx = blockIdx.x * blockDim.x + threadIdx.x;
  if (idx < NNODES * FDIM) rst[idx] = 0.f;
  if (idx < NNODES * NH) {
    emax[idx] = (int)0x80000000;
    denom[idx] = 0.f;
  }
}

// e = leaky_relu(el[src] + er[dst]); segment-max via ordered-int atomicMax.
__global__ void edge_logits_kernel(const int* __restrict__ src, const int* __restrict__ dst,
                                   const float* __restrict__ el, const float* __restrict__ er,
                                   float* __restrict__ ebuf, int* __restrict__ emax) {
  const int e = blockIdx.x * blockDim.x + threadIdx.x;
  if (e >= NEDGES) return;
  const int s = src[e], d = dst[e];
  #pragma unroll
  for (int h = 0; h < NH; ++h) {
    float v = el[s * NH + h] + er[d * NH + h];
    v = v > 0.f ? v : NEG_SLOPE * v;
    ebuf[(size_t)e * NH + h] = v;
    atomicMax(&emax[d * NH + h], f2ord(v));
  }
}

// ex = exp(e - emax[dst]); denom[dst] += ex (segment sum).
__global__ void edge_exp_kernel(const int* __restrict__ dst, const int* __restrict__ emax,
                                float* __restrict__ ebuf, float* __restrict__ denom) {
  const int e = blockIdx.x * blockDim.x + threadIdx.x;
  if (e >= NEDGES) return;
  const int d = dst[e];
  #pragma unroll
  for (int h = 0; h < NH; ++h) {
    float ex = expf(ebuf[(size_t)e * NH + h] - ord2f(emax[d * NH + h]));
    ebuf[(size_t)e * NH + h] = ex;
    atomicAdd(&denom[d * NH + h], ex);
  }
}

// rst[dst] += (ex/denom[dst]) * feat[src]
// One wave per edge: 3 x b128 gather per lane + 12 global_atomic_add_f32.
__global__ __launch_bounds__(256) void edge_aggregate_kernel(const int* __restrict__ src,
                                                             const int* __restrict__ dst,
                                                             const float* __restrict__ ebuf,
                                                             const float* __restrict__ denom,
                                                             const float* __restrict__ feat,
                                                             float* __restrict__ rst) {
  const int e = blockIdx.x * 8 + (threadIdx.x >> 5);
  if (e >= NEDGES) return;
  const int lane = threadIdx.x & 31;
  const int s = src[e], d = dst[e];
  const float a0 = ebuf[(size_t)e * NH + 0] / denom[d * NH + 0];
  const float a1 = ebuf[(size_t)e * NH + 1] / denom[d * NH + 1];
  const float a2 = ebuf[(size_t)e * NH + 2] / denom[d * NH + 2];
  const float4* fs4 = (const float4*)(feat + (size_t)s * FDIM);
  float* rd = rst + (size_t)d * FDIM;
  #pragma unroll
  for (int it = 0; it < 3; ++it) {
    const int i = it * 32 + lane;               // float4 index, 0..95
    const float4 v = fs4[i];
    const float alpha = (i < 32) ? a0 : ((i < 64) ? a1 : a2);
    atomicAdd(&rd[i * 4 + 0], alpha * v.x);
    atomicAdd(&rd[i * 4 + 1], alpha * v.y);
    atomicAdd(&rd[i * 4 + 2], alpha * v.z);
    atomicAdd(&rd[i * 4 + 3], alpha * v.w);
  }
}

// h[n][d] = mean over heads of (rst[n][h][d] + bias[h][d])
__global__ void finalize_kernel(const float* __restrict__ rst, const float* __restrict__ bias,
                                float* __restrict__ hout) {
  const int idx = blockIdx.x * blockDim.x + threadIdx.x;
  if (idx >= NNODES * HD) return;
  const int n = idx >> 7, d = idx & 127;
  const size_t base = (size_t)n * FDIM;
  float v = (rst[base + d] + bias[d]) + (rst[base + HD + d] + bias[HD + d]) +
            (rst[base + 2 * HD + d] + bias[2 * HD + d]);
  hout[idx] = v * (1.0f / 3.0f);
}

// ---------------------------------------------------------------------------
// Final gather + concat:  seq [5120 x 512], seq_r [5120 x 384], s_hist_t [5120]
// ---------------------------------------------------------------------------
__global__ __launch_bounds__(128) void concat_kernel(const float* __restrict__ h,
                                                     const float* __restrict__ ent_embeds,
                                                     const float* __restrict__ rel_embeds,
                                                     const float* __restrict__ glob,
                                                     const float* __restrict__ s_hist_t,
                                                     const int* __restrict__ node_ids,
                                                     const int* __restrict__ s_tem,
                                                     const int* __restrict__ r_tem,
                                                     float* __restrict__ out) {
  const int t = blockIdx.x;          // 0 .. B*L-1
  const int d = threadIdx.x;         // 0 .. 127
  const int b = t / LSEQ;
  const float emb = h[(size_t)node_ids[t] * HD + d];
  const float ent = ent_embeds[(size_t)s_tem[b] * HD + d];
  const float rel = rel_embeds[(size_t)r_tem[b] * HD + d];
  const float g   = glob[(size_t)t * HD + d];

  float* seq = out;                                        // [5120][512]
  seq[(size_t)t * 512 + 0 * HD + d] = emb;
  seq[(size_t)t * 512 + 1 * HD + d] = ent;
  seq[(size_t)t * 512 + 2 * HD + d] = rel;
  seq[(size_t)t * 512 + 3 * HD + d] = g;

  float* seqr = out + (size_t)BBATCH * LSEQ * 512;         // [5120][384]
  seqr[(size_t)t * 384 + 0 * HD + d] = emb;
  seqr[(size_t)t * 384 + 1 * HD + d] = ent;
  seqr[(size_t)t * 384 + 2 * HD + d] = g;

  if (d == 0) {
    float* out3 = out + (size_t)BBATCH * LSEQ * (512 + 384);
    out3[t] = s_hist_t[t];
  }
}

extern "C" void kernel_launch(void* const* d_in, const int* in_sizes, int n_in,
                              void* d_out, int out_size, void* d_ws, size_t ws_size,
                              hipStream_t stream) {
  const float* node_feat = (const float*)d_in[0];
  const float* W1        = (const float*)d_in[1];
  const float* attn_l1   = (const float*)d_in[2];
  const float* attn_r1   = (const float*)d_in[3];
  const float* b1        = (const float*)d_in[4];
  const float* W2        = (const float*)d_in[5];
  const float* attn_l2   = (const float*)d_in[6];
  const float* attn_r2   = (const float*)d_in[7];
  const float* b2        = (const float*)d_in[8];
  const float* ent_emb   = (const float*)d_in[9];
  const float* rel_emb   = (const float*)d_in[10];
  const float* glob      = (const float*)d_in[11];
  const float* s_hist_t  = (const float*)d_in[12];
  const int*   edge_src  = (const int*)d_in[13];
  const int*   edge_dst  = (const int*)d_in[14];
  const int*   node_ids  = (const int*)d_in[15];
  const int*   s_tem     = (const int*)d_in[16];
  const int*   r_tem     = (const int*)d_in[17];

  float* ws    = (float*)d_ws;
  float* feat  = ws;                               // N*384
  float* rst   = feat  + (size_t)NNODES * FDIM;    // N*384
  float* hbuf  = rst   + (size_t)NNODES * FDIM;    // N*128
  float* el    = hbuf  + (size_t)NNODES * HD;      // N*3
  float* er    = el    + (size_t)NNODES * NH;      // N*3
  int*   emax  = (int*)(er + (size_t)NNODES * NH); // N*3 (ordered-int)
  float* denom = (float*)(emax + (size_t)NNODES * NH); // N*3
  float* ebuf  = denom + (size_t)NNODES * NH;      // E*3

  float* out = (float*)d_out;

  auto run_layer = [&](const float* xin, const float* W, const float* al,
                       const float* ar, const float* bias) {
    layer_init_kernel<<<dim3((NNODES * FDIM + 255) / 256), dim3(256), 0, stream>>>(
        rst, emax, denom);
    gemm_wmma_kernel<<<dim3(NNODES / 16), dim3(256), 0, stream>>>(xin, W, feat);
    attn_dots_kernel<<<dim3(NNODES / 4), dim3(128), 0, stream>>>(feat, al, ar, el, er);
    edge_logits_kernel<<<dim3((NEDGES + 255) / 256), dim3(256), 0, stream>>>(
        edge_src, edge_dst, el, er, ebuf, emax);
    edge_exp_kernel<<<dim3((NEDGES + 255) / 256), dim3(256), 0, stream>>>(
        edge_dst, emax, ebuf, denom);
    edge_aggregate_kernel<<<dim3(NEDGES / 8), dim3(256), 0, stream>>>(
        edge_src, edge_dst, ebuf, denom, feat, rst);
    finalize_kernel<<<dim3((NNODES * HD) / 256), dim3(256), 0, stream>>>(rst, bias, hbuf);
  };

  run_layer(node_feat, W1, attn_l1, attn_r1, b1);
  run_layer(hbuf,      W2, attn_l2, attn_r2, b2);

  concat_kernel<<<dim3(BBATCH * LSEQ), dim3(HD), 0, stream>>>(
      hbuf, ent_emb, rel_emb, glob, s_hist_t, node_ids, s_tem, r_tem, out);
}